// TTTLayer_7499012899548
// MI455X (gfx1250) — compile-verified
//
#include <hip/hip_runtime.h>
#include <cstdint>

typedef float v2f __attribute__((ext_vector_type(2)));
typedef float v8f __attribute__((ext_vector_type(8)));

#define INNER_LR 0.01f

#define B_  4
#define S_  2048
#define D_  1024
#define DS_ 256

#define JC  16   // W columns owned per workgroup (16 slices per batch -> 64 WGs)
#define CH  16   // chunk length (timesteps folded into WMMA GEMMs)
#define NCH (S_ / CH)

__device__ __forceinline__ v8f wmma_f32(v2f a, v2f b, v8f c) {
  // V_WMMA_F32_16X16X4_F32 : D(16x16 f32) = A(16x4 f32) * B(4x16 f32) + C
  return __builtin_amdgcn_wmma_f32_16x16x4_f32(false, a, false, b, (short)0, c,
                                               false, false);
}

// CDNA5 async global->LDS copy (ASYNCcnt-tracked), GV addressing mode.
__device__ __forceinline__ void async_load_b128(unsigned lds_off,
                                                const float* gaddr) {
  asm volatile("global_load_async_to_lds_b128 %0, %1, off"
               :: "v"(lds_off), "v"(gaddr)
               : "memory");
}
__device__ __forceinline__ void wait_async0() {
  asm volatile("s_wait_asynccnt 0x0" ::: "memory");
}

// ---------------------------------------------------------------------------
// fp32 WMMA GEMM: C[M,N] = A[M,K]*B[K,N] + bias[N] (+ resid[M,N])
// Wave tile 16(M) x 64(N); software-pipelined (prefetch distance 1).
// ---------------------------------------------------------------------------
__global__ void gemm16x64_f32(const float* __restrict__ A,
                              const float* __restrict__ Bm,
                              const float* __restrict__ bias,
                              const float* __restrict__ resid,
                              float* __restrict__ C,
                              int M, int N, int K) {
  const int lane  = threadIdx.x & 31;
  const int wid   = blockIdx.x * 8 + (threadIdx.x >> 5);
  const int tilesN = N >> 6;
  const int row0  = (wid / tilesN) << 4;
  const int col0  = (wid % tilesN) << 6;
  const int mrow  = lane & 15;
  const int khi   = (lane >> 4) << 1;
  const int hi    = lane >> 4;

  v8f acc0 = {}, acc1 = {}, acc2 = {}, acc3 = {};
  const float* Ap = A + (size_t)(row0 + mrow) * K + khi;
  const float* Bp = Bm + (size_t)khi * N + col0 + mrow;
  const size_t bstep = (size_t)4 * N;

  // prologue fragments (k = 0)
  v2f a_c = *(const v2f*)Ap;
  v2f b0, b1, b2, b3;
  b0.x = Bp[0];  b0.y = Bp[N];
  b1.x = Bp[16]; b1.y = Bp[N + 16];
  b2.x = Bp[32]; b2.y = Bp[N + 32];
  b3.x = Bp[48]; b3.y = Bp[N + 48];

#pragma unroll 2
  for (int k = 4; k < K; k += 4) {
    Ap += 4; Bp += bstep;
    v2f a_n = *(const v2f*)Ap;                 // prefetch next k-step
    v2f n0, n1, n2, n3;
    n0.x = Bp[0];  n0.y = Bp[N];
    n1.x = Bp[16]; n1.y = Bp[N + 16];
    n2.x = Bp[32]; n2.y = Bp[N + 32];
    n3.x = Bp[48]; n3.y = Bp[N + 48];
    acc0 = wmma_f32(a_c, b0, acc0);            // compute current k-step
    acc1 = wmma_f32(a_c, b1, acc1);
    acc2 = wmma_f32(a_c, b2, acc2);
    acc3 = wmma_f32(a_c, b3, acc3);
    a_c = a_n; b0 = n0; b1 = n1; b2 = n2; b3 = n3;
  }
  acc0 = wmma_f32(a_c, b0, acc0);              // epilogue k-step
  acc1 = wmma_f32(a_c, b1, acc1);
  acc2 = wmma_f32(a_c, b2, acc2);
  acc3 = wmma_f32(a_c, b3, acc3);

  const float bs0 = bias[col0 + mrow];
  const float bs1 = bias[col0 + mrow + 16];
  const float bs2 = bias[col0 + mrow + 32];
  const float bs3 = bias[col0 + mrow + 48];
#pragma unroll
  for (int r = 0; r < 8; ++r) {
    const int m = row0 + r + (hi << 3);
    const size_t base = (size_t)m * N + col0 + mrow;
    float r0 = 0.f, r1 = 0.f, r2 = 0.f, r3 = 0.f;
    if (resid) { r0 = resid[base];      r1 = resid[base + 16];
                 r2 = resid[base + 32]; r3 = resid[base + 48]; }
    C[base]      = acc0[r] + bs0 + r0;
    C[base + 16] = acc1[r] + bs1 + r1;
    C[base + 32] = acc2[r] + bs2 + r2;
    C[base + 48] = acc3[r] + bs3 + r3;
  }
}

// ---------------------------------------------------------------------------
// TTT scan, exact chunked (WY) form, column-sharded fast weights.
// grid = B_ * (DS_/JC) = 64 blocks, 256 threads (8 waves).
// Per chunk of CH=16 steps (state W0 = Ws, slice of JC columns):
//   P0 = F * Ws        (waves 0..3, K split 4 ways, WMMA)
//   G  = F * F^T       (waves 4..7, K split 4 ways, WMMA)
//   combine partials ; triangular solve E ; preds = E + Fslice
//   Ws -= lr * F^T * E (all 8 waves, WMMA)
// F chunks are double-buffered in LDS and prefetched with
// global_load_async_to_lds_b128 (ASYNCcnt) one chunk ahead.
// ---------------------------------------------------------------------------
__global__ void ttt_scan(const float* __restrict__ feat,
                         float* __restrict__ preds) {
  __shared__ float Ws[DS_ * JC];       // 16 KB : W[:, cb..cb+JC)
  __shared__ float Fc[2][CH * DS_];    // 32 KB : double-buffered F chunk
  __shared__ float Pp[4][CH * JC];     //  4 KB : P0 K-partials
  __shared__ float Gp[4][CH * CH];     //  4 KB : Gram K-partials
  __shared__ float Ec[CH * JC];        //  1 KB : rhs -> errors (in place)
  __shared__ float Gm[CH * CH];        //  1 KB : Gram

  const int tid  = threadIdx.x;
  const int lane = tid & 31;
  const int w    = tid >> 5;
  const int b    = blockIdx.x >> 4;             // batch
  const int cb   = (blockIdx.x & 15) * JC;      // column slice base
  const int mrow = lane & 15;
  const int khi  = (lane >> 4) << 1;
  const int hi   = lane >> 4;

  for (int i = tid; i < DS_ * JC; i += 256) Ws[i] = 0.f;   // W0 = 0

  // synchronous load of chunk 0 into buffer 0
#pragma unroll
  for (int t = 0; t < CH; ++t)
    Fc[0][t * DS_ + tid] = feat[((size_t)b * S_ + t) * DS_ + tid];

  for (int c = 0; c < NCH; ++c) {
    const int cur = c & 1;
    wait_async0();          // prefetch of Fc[cur] (issued last chunk) done
    __syncthreads();

    // ---- issue async prefetch of chunk c+1 into the other buffer ----
    if (c + 1 < NCH) {
      const float* src =
          feat + ((size_t)b * S_ + (size_t)(c + 1) * CH) * DS_;
      const unsigned dstbase =
          (unsigned)(uintptr_t)&Fc[cur ^ 1][0] + (unsigned)(tid * 16);
#pragma unroll
      for (int p = 0; p < 4; ++p)                 // 16 KB contiguous span
        async_load_b128(dstbase + p * 4096, src + tid * 4 + p * 1024);
    }

    const float* F = Fc[cur];

    // ---- phase 2: P0 partials (waves 0..3) / Gram partials (waves 4..7) --
    if (w < 4) {
      const int kbase = w << 6;                   // 64 K-values each
      v8f acc = {};
      for (int k = kbase; k < kbase + 64; k += 4) {
        v2f a, bf;
        a.x  = F[mrow * DS_ + k + khi];
        a.y  = F[mrow * DS_ + k + khi + 1];
        bf.x = Ws[(k + khi) * JC + mrow];
        bf.y = Ws[(k + khi + 1) * JC + mrow];
        acc = wmma_f32(a, bf, acc);
      }
#pragma unroll
      for (int r = 0; r < 8; ++r)
        Pp[w][(r + (hi << 3)) * JC + mrow] = acc[r];
    } else {
      const int g = w - 4;
      const int kbase = g << 6;
      v8f acc = {};
      for (int k = kbase; k < kbase + 64; k += 4) {
        v2f a;
        a.x = F[mrow * DS_ + k + khi];
        a.y = F[mrow * DS_ + k + khi + 1];
        acc = wmma_f32(a, a, acc);                // B-frag == A-frag (F F^T)
      }
#pragma unroll
      for (int r = 0; r < 8; ++r)
        Gp[g][(r + (hi << 3)) * CH + mrow] = acc[r];
    }
    __syncthreads();

    // ---- combine partials: rhs = P0 - Fslice ; Gm = sum Gp ----
    {
      const int t = tid >> 4, j = tid & 15;
      Ec[tid] = (Pp[0][tid] + Pp[1][tid] + Pp[2][tid] + Pp[3][tid])
                - F[t * DS_ + cb + j];
      Gm[tid] = Gp[0][tid] + Gp[1][tid] + Gp[2][tid] + Gp[3][tid];
    }
    __syncthreads();

    // ---- triangular solve (in place): E[t] = rhs[t] - lr*sum G[t,u]E[u] --
    if (tid < JC) {
      float e[CH];
#pragma unroll
      for (int t = 0; t < CH; ++t) {
        float v = Ec[t * JC + tid];
        for (int u = 0; u < t; ++u)
          v -= INNER_LR * Gm[t * CH + u] * e[u];
        e[t] = v;
        Ec[t * JC + tid] = v;
        preds[((size_t)b * S_ + (size_t)c * CH + t) * DS_ + cb + tid] =
            v + F[t * DS_ + cb + tid];            // pred = err + f
      }
    }
    __syncthreads();

    // ---- Ws -= lr * F^T * E : (256x16)*(16x16), 16 tiles / 8 waves ----
#pragma unroll
    for (int q = 0; q < 2; ++q) {
      const int mt = (w << 1) + q;                // 16-row block of W
      const int m  = (mt << 4) + mrow;
      v8f acc = {};
#pragma unroll
      for (int k = 0; k < CH; k += 4) {
        v2f a, bf;
        a.x  = F[(k + khi) * DS_ + m];            // F^T[m][k] = F[k][m]
        a.y  = F[(k + khi + 1) * DS_ + m];
        bf.x = Ec[(k + khi) * JC + mrow];
        bf.y = Ec[(k + khi + 1) * JC + mrow];
        acc = wmma_f32(a, bf, acc);
      }
#pragma unroll
      for (int r = 0; r < 8; ++r) {
        const int mi = (mt << 4) + r + (hi << 3);
        Ws[mi * JC + mrow] -= INNER_LR * acc[r];
      }
    }
    __syncthreads();
  }
}

// ---------------------------------------------------------------------------
extern "C" void kernel_launch(void* const* d_in, const int* in_sizes, int n_in,
                              void* d_out, int out_size, void* d_ws, size_t ws_size,
                              hipStream_t stream) {
  const float* x  = (const float*)d_in[0];
  const float* Wd = (const float*)d_in[1];
  const float* bd = (const float*)d_in[2];
  const float* Wu = (const float*)d_in[3];
  const float* bu = (const float*)d_in[4];
  float* out  = (float*)d_out;

  float* feat  = (float*)d_ws;                       //  8 MB
  float* preds = feat + (size_t)B_ * S_ * DS_;       //  8 MB

  // K1: feat = x @ W_down + b_down   (M=8192, N=256, K=1024)
  {
    const int M = B_ * S_, N = DS_, K = D_;
    const int waves = (M / 16) * (N / 64);
    gemm16x64_f32<<<waves / 8, 256, 0, stream>>>(x, Wd, bd, nullptr, feat, M, N, K);
  }

  // K2: sequential fast-weight scan (chunked WMMA form, async F prefetch)
  ttt_scan<<<B_ * (DS_ / JC), 256, 0, stream>>>(feat, preds);

  // K3: out = preds @ W_up + b_up + x   (M=8192, N=1024, K=256)
  {
    const int M = B_ * S_, N = D_, K = DS_;
    const int waves = (M / 16) * (N / 64);
    gemm16x64_f32<<<waves / 8, 256, 0, stream>>>(preds, Wu, bu, x, out, M, N, K);
  }
}